// HyperbolicSelfAttention_40243843564010
// MI455X (gfx1250) — compile-verified
//
#include <hip/hip_runtime.h>
#include <stdint.h>

typedef __attribute__((ext_vector_type(2))) float v2f;
typedef __attribute__((ext_vector_type(8))) float v8f;

#define DD   1024
#define BATCH 32

// ---------- fast scalar helpers (gfx1250 VALU transcendentals) ----------
__device__ __forceinline__ float fast_rcp(float x) { return __builtin_amdgcn_rcpf(x); }
// v_log_f32 is log2; convert to natural log
__device__ __forceinline__ float fast_ln(float x)  { return 0.6931471805599453f * __builtin_amdgcn_logf(x); }

__device__ __forceinline__ float artanh_clip(float x) {
    x = fminf(x,  1.0f - 1e-7f);
    x = fmaxf(x, -1.0f + 1e-7f);
    return 0.5f * fast_ln((1.0f + x) * fast_rcp(1.0f - x));
}

// res = tanh(|mx|/|x| * artanh(|x|)) * mx/|mx|   (scalar 1-D ball mobius_matvec tail)
__device__ __forceinline__ float mobius_epilogue(float mx, float x) {
    float xn = fmaxf(fabsf(x),  1e-15f);
    float mn = fmaxf(fabsf(mx), 1e-15f);
    float t  = tanhf(mn * fast_rcp(xn) * artanh_clip(xn));
    return t * mx * fast_rcp(mn);
}

// ---------- CDNA5 async copy: global -> LDS (ASYNCcnt-tracked DMA path) ----------
// GLOBAL_LOAD_ASYNC_TO_LDS_B32: VDST = per-lane LDS byte address, VADDR = 64-bit
// global address, no SADDR ("off"). LDS offset = low 32 bits of flat shared ptr
// (ISA 10.2: LDS aperture truncates to addr[31:0]).
__device__ __forceinline__ void async_copy_f32(const float* gsrc, float* lds_dst) {
    uint32_t laddr = (uint32_t)(uintptr_t)lds_dst;
    asm volatile("global_load_async_to_lds_b32 %0, %1, off"
                 :: "v"(laddr), "v"((uint64_t)(uintptr_t)gsrc)
                 : "memory");
}
__device__ __forceinline__ void wait_async0() {
    asm volatile("s_wait_asynccnt 0x0" ::: "memory");
}

// ---------- Kernel 1: fused QKV GEMM (f32 WMMA) + hyperbolic epilogues ----------
// grid: 48 blocks x 128 threads = 192 waves; wave -> (proj in {Q,K,V}, 16-col tile)
// Each wave: 32(M) x 16(N) output tile, K-loop over 1024 in steps of 4
// using V_WMMA_F32_16X16X4_F32 (A 16x4, B 4x16, C/D 16x16 f32).
__global__ __launch_bounds__(128) void qkv_wmma_kernel(
    const float* __restrict__ X,
    const float* __restrict__ Wq,
    const float* __restrict__ Wk,
    const float* __restrict__ Wv,
    float* __restrict__ Qs,
    float* __restrict__ Ks,
    float* __restrict__ Lv)
{
    int wave = (blockIdx.x << 2) + (threadIdx.x >> 5);   // 0..191
    int proj = wave >> 6;                                // 0:Q 1:K 2:V
    int n0   = (wave & 63) << 4;                         // output-col tile base
    const float* W = (proj == 0) ? Wq : (proj == 1) ? Wk : Wv;

    int lane = threadIdx.x & 31;
    int l15  = lane & 15;
    int hi   = lane >> 4;          // half-wave selects K pair
    int kk   = hi << 1;

    // A (16x4 f32): lane l15 = row M, VGPR v = K = v + 2*hi  -> float2 load
    // B (4x16 f32): lane l15 = col N, VGPR v = K = v + 2*hi; B[k][n] = W[n][k] -> float2 from W row n
    const float* arow0 = X + l15 * DD;          // batch rows 0..15
    const float* arow1 = X + (l15 + 16) * DD;   // batch rows 16..31
    const float* brow  = W + (n0 + l15) * DD;

    v8f c0 = {};   // batch rows 0..15
    v8f c1 = {};   // batch rows 16..31
    for (int k = 0; k < DD; k += 4) {
        if ((k & 63) == 0) {
            // stream-ahead prefetch of the weight row (lowers to global_prefetch_b8)
            __builtin_prefetch(brow + k + 64, 0, 1);
        }
        v2f a0 = *(const v2f*)(arow0 + k + kk);
        v2f a1 = *(const v2f*)(arow1 + k + kk);
        v2f b  = *(const v2f*)(brow  + k + kk);
        c0 = __builtin_amdgcn_wmma_f32_16x16x4_f32(false, a0, false, b, (short)0, c0, false, false);
        c1 = __builtin_amdgcn_wmma_f32_16x16x4_f32(false, a1, false, b, (short)0, c1, false, false);
    }

    // C/D layout: VGPR r, lanes0-15: M=r, lanes16-31: M=r+8; N = l15
    int col = n0 + l15;
    #pragma unroll
    for (int r = 0; r < 8; ++r) {
        int b0 = r + (hi << 3);        // batch row for c0
        int b1 = b0 + 16;              // batch row for c1
        float r0 = mobius_epilogue(c0[r], X[b0 * DD + col]);
        float r1 = mobius_epilogue(c1[r], X[b1 * DD + col]);
        if (proj == 0) {
            Qs[b0 * DD + col] = r0;
            Qs[b1 * DD + col] = r1;
        } else if (proj == 1) {
            Ks[b0 * DD + col] = r0;
            Ks[b1 * DD + col] = r1;
        } else {
            // projx (ball eps 4e-3) then logmap0: sign(v)*artanh(|v|)
            const float maxn = 1.0f - 4e-3f;
            float v0 = (fabsf(r0) > maxn) ? copysignf(maxn, r0) : r0;
            float v1 = (fabsf(r1) > maxn) ? copysignf(maxn, r1) : r1;
            float nv0 = fmaxf(fabsf(v0), 1e-15f);
            float nv1 = fmaxf(fabsf(v1), 1e-15f);
            Lv[b0 * DD + col] = artanh_clip(nv0) * v0 * fast_rcp(nv0);
            Lv[b1 * DD + col] = artanh_clip(nv1) * v1 * fast_rcp(nv1);
        }
    }
}

// ---------- Kernel 2: pairwise Poincare attention (transcendental-bound) ----------
// grid 128 x 256: block -> (batch b, 256-wide chunk of i).
// K and Lv rows staged into LDS via CDNA5 async global->LDS DMA.
__global__ __launch_bounds__(256) void attn_kernel(
    const float* __restrict__ Qs,
    const float* __restrict__ Ks,
    const float* __restrict__ Lv,
    float* __restrict__ out)
{
    __shared__ float sK[DD];
    __shared__ float sK2[DD];
    __shared__ float sL[DD];

    int b   = blockIdx.x >> 2;
    int tid = threadIdx.x;

    // Async-stage K[b,:] and Lv[b,:] into LDS (ASYNCcnt path, no VGPR round-trip)
    for (int j = tid; j < DD; j += 256) {
        async_copy_f32(Ks + b * DD + j, &sK[j]);
        async_copy_f32(Lv + b * DD + j, &sL[j]);
    }
    wait_async0();          // this wave's async transfers complete
    __syncthreads();        // all waves' transfers visible

    // derive K^2 in LDS
    for (int j = tid; j < DD; j += 256) {
        float k = sK[j];
        sK2[j] = k * k;
    }
    __syncthreads();

    int i    = ((blockIdx.x & 3) << 8) + tid;
    float q  = Qs[b * DD + i];
    float q2 = q * q;

    float acc = 0.0f;
    #pragma unroll 4
    for (int j = 0; j < DD; ++j) {
        float k  = sK[j];
        float k2 = sK2[j];
        // mobius_add(-q, k):
        //   num = (1 - 2qk + k^2)(-q) + (1 - q^2)k
        //   den = max(1 - 2qk + q^2 k^2, 1e-15)
        float a   = 1.0f - 2.0f * q * k;
        float num = (1.0f - q2) * k - (a + k2) * q;
        float den = fmaxf(a + q2 * k2, 1e-15f);
        float d   = fmaxf(fabsf(num) * fast_rcp(den), 1e-15f);   // norm clamp
        float t   = fminf(d, 1.0f - 1e-7f);                      // artanh clip (d >= 0)
        // dist = 2*artanh(t) = ln((1+t)/(1-t))
        float dist = fast_ln((1.0f + t) * fast_rcp(1.0f - t));
        acc += sL[j] * fast_rcp(dist + 1e-5f);
    }
    // expmap0 for scalar: tanh(w)
    out[b * DD + i] = tanhf(acc);
}

// ---------- launch ----------
extern "C" void kernel_launch(void* const* d_in, const int* in_sizes, int n_in,
                              void* d_out, int out_size, void* d_ws, size_t ws_size,
                              hipStream_t stream) {
    const float* X  = (const float*)d_in[0];
    const float* Wq = (const float*)d_in[1];
    const float* Wk = (const float*)d_in[2];
    const float* Wv = (const float*)d_in[3];
    float* out = (float*)d_out;

    float* Qs = (float*)d_ws;                 // 32*1024 f32
    float* Ks = Qs + BATCH * DD;              // 32*1024 f32
    float* Lv = Ks + BATCH * DD;              // 32*1024 f32  (384 KB total)

    qkv_wmma_kernel<<<48, 128, 0, stream>>>(X, Wq, Wk, Wv, Qs, Ks, Lv);
    attn_kernel<<<128, 256, 0, stream>>>(Qs, Ks, Lv, out);
}